// DMCustom_28338194219111
// MI455X (gfx1250) — compile-verified
//
#include <hip/hip_runtime.h>

// Reference: DDPM pixel-swap "cold diffusion" degrade.
//   perm = compose 4096 transpositions (i, q[i]) sequentially on identity
//   out[b, loc] = x[b, perm[loc]],  x: (8192, 1, 64, 64) f32
//
// Kernel A (1 block): build perm.
//   - chunk the 4096 swaps into 16 chunks of 256; 16 lanes build the 16
//     chunk-permutations concurrently (serial chain 4096 -> 256 steps),
//     using ~278 KB of the WGP's 320 KB LDS (CDNA5-only capacity).
//   - fold P <- P o sigma_c with parallel 4096-wide gathers.
// Kernel B (1024 blocks): memory-bound gather, 268 MB total -> ~11.5 us at
//   23.3 TB/s. perm staged into LDS per block via CDNA5 async global->LDS
//   (global_load_async_to_lds_b32 + s_wait_asynccnt). Streaming x/out use
//   non-temporal hints so the reused 16 KB perm table stays in L2;
//   ds_load_b128 for perm indices; coalesced b128 NT stores for out.

#define HW      4096
#define NCHUNK  16
#define CHUNK   256       // NCHUNK * CHUNK == HW
#define SROW    4097      // padded row stride (words) to stagger LDS banks
#define RPB     8         // rows per block in gather kernel

typedef float f32x4 __attribute__((ext_vector_type(4)));  // native vector for NT store

// beta = (BETA2-BETA1)*(t/1000) + BETA1 with XLA's separate mul/add rounding:
// forbid mul+add -> fma contraction so we match the reference bit-for-bit.
__device__ __forceinline__ float beta_of_t(float tf)
{
#pragma clang fp contract(off)
    const float prod = (float)(0.02 - 1e-4) * (tf / 1000.0f);
    return prod + 1e-4f;
}

__global__ __launch_bounds__(512) void build_perm_kernel(
    const float* __restrict__ u1, const float* __restrict__ u2,
    const int* __restrict__ tptr, int* __restrict__ perm_out)
{
    __shared__ int q[HW];              // 16 KB: swap partner per step
    __shared__ int sig[NCHUNK * SROW]; // ~262 KB: per-chunk permutations

    const int tid = threadIdx.x;

    // ---- phase 1: q[i] and identity init (parallel) ----
    const float beta = beta_of_t((float)tptr[0]);

    for (int i = tid; i < HW; i += 512) {
        const int r = i >> 6, c = i & 63;
        // match JAX: left-assoc f32 multiply chain, trunc-toward-zero cast
        const int d1 = (int)(((u1[i] - 0.5f) * 2.0f) * beta * 64.0f); // -> cols
        const int d2 = (int)(((u2[i] - 0.5f) * 2.0f) * beta * 64.0f); // -> rows
        const int tr = (r + d2) & 63;   // == python-mod 64 (power of two)
        const int tc = (c + d1) & 63;
        q[i] = (tr << 6) | tc;
    }
    for (int c = 0; c < NCHUNK; ++c)
        for (int x = tid; x < HW; x += 512)
            sig[c * SROW + x] = x;      // identity
    __syncthreads();

    // ---- phase 2: 16 lanes each serially apply their chunk's 256 swaps ----
    // Identical entry-swap semantics to the reference scan, per chunk.
    if (tid < NCHUNK) {
        int* s = &sig[tid * SROW];
        const int base = tid * CHUNK;
        for (int j = 0; j < CHUNK; ++j) {
            const int pi = base + j;
            const int qi = q[pi];
            const int vp = s[pi];
            const int vq = s[qi];
            s[pi] = vq;
            s[qi] = vp;
        }
    }
    __syncthreads();

    // ---- phase 3: fold P <- P o sigma_c (parallel gathers, double buffer) ----
    int* A = &sig[0];
    int* B = q;                          // q is dead after phase 2; reuse
    for (int c = 1; c < NCHUNK; ++c) {
        const int* S = &sig[c * SROW];
        for (int x = tid; x < HW; x += 512)
            B[x] = A[S[x]];              // (P o sigma_c)[x] = P[sigma_c[x]]
        __syncthreads();
        int* t2 = A; A = B; B = t2;
    }

    // ---- phase 4: publish perm ----
    for (int x = tid; x < HW; x += 512)
        perm_out[x] = A[x];
}

__global__ __launch_bounds__(256) void gather_kernel(
    const float* __restrict__ x, const int* __restrict__ perm,
    float* __restrict__ out)
{
    __shared__ int s_perm[HW];
    const int tid = threadIdx.x;

    // CDNA5 async copy global -> LDS (GVS mode: saddr pair + 32-bit voffset),
    // tracked by ASYNCcnt. 16 outstanding per wave, then drain + barrier.
    for (int k = tid; k < HW; k += 256) {
        unsigned lds_addr = (unsigned)(unsigned long long)(&s_perm[k]); // low 32b = LDS offset
        unsigned goff     = (unsigned)(k * 4);
        asm volatile("global_load_async_to_lds_b32 %0, %1, %2"
                     :: "v"(lds_addr), "v"(goff), "s"(perm) : "memory");
    }
    asm volatile("s_wait_asynccnt 0x0" ::: "memory");
    __syncthreads();

    const long long row0 = (long long)blockIdx.x * RPB;
    for (int r = 0; r < RPB; ++r) {
        const float* __restrict__ xr   = x   + (row0 + r) * HW;
        float*       __restrict__ outr = out + (row0 + r) * HW;
#pragma unroll
        for (int g = 0; g < 4; ++g) {
            const int c0 = (tid + g * 256) * 4;   // 4 consecutive cols/thread
            // one ds_load_b128 for 4 perm indices (16B aligned)
            const int4 p4 = *(const int4*)&s_perm[c0];
            f32x4 v;
            // streamed exactly once -> non-temporal, don't thrash L2
            v.x = __builtin_nontemporal_load(&xr[p4.x]);
            v.y = __builtin_nontemporal_load(&xr[p4.y]);
            v.z = __builtin_nontemporal_load(&xr[p4.z]);
            v.w = __builtin_nontemporal_load(&xr[p4.w]);
            __builtin_nontemporal_store(v, (f32x4*)(outr + c0)); // b128 NT store
        }
    }
}

extern "C" void kernel_launch(void* const* d_in, const int* in_sizes, int n_in,
                              void* d_out, int out_size, void* d_ws, size_t ws_size,
                              hipStream_t stream)
{
    const float* x  = (const float*)d_in[0];   // (8192,1,64,64)
    const float* u1 = (const float*)d_in[1];   // (64,64)
    const float* u2 = (const float*)d_in[2];   // (64,64)
    const int*   t  = (const int*)d_in[3];     // scalar
    float* out = (float*)d_out;
    int* perm  = (int*)d_ws;                   // 4096 ints of scratch

    build_perm_kernel<<<1, 512, 0, stream>>>(u1, u2, t, perm);
    gather_kernel<<<8192 / RPB, 256, 0, stream>>>(x, perm, out);
}